// PushUp_6906307412023
// MI455X (gfx1250) — compile-verified
//
#include <hip/hip_runtime.h>
#include <stdint.h>

// Problem constants (from reference): N=500000, F=64, K=5, N_UP=100000
#define F_DIM 64
#define K_DIM 5
#define TILE_ROWS 16   // rows of `features` per TDM tile (16*64*4B = 4KB)

typedef __attribute__((ext_vector_type(4))) unsigned int u32x4;
typedef __attribute__((ext_vector_type(8))) int          i32x8;
typedef __attribute__((ext_vector_type(4))) int          i32x4;

// ---------------------------------------------------------------------------
// Kernel 1: zero the accumulator workspace (denom[N] + feat_acc[N*64]).
// Must run every launch: graph replay does not re-poison d_ws.
// ---------------------------------------------------------------------------
__global__ void zero_ws_kernel(float* __restrict__ p, long long n) {
  long long stride = (long long)gridDim.x * blockDim.x * 4;
  for (long long i = ((long long)blockIdx.x * blockDim.x + threadIdx.x) * 4;
       i < n; i += stride) {
    if (i + 3 < n) {
      *(float4*)(p + i) = make_float4(0.f, 0.f, 0.f, 0.f);
    } else {
      for (long long j = i; j < n; ++j) p[j] = 0.f;
    }
  }
}

// ---------------------------------------------------------------------------
// TDM issue: DMA one TILE_ROWS x 64 fp32 tile of `features` into LDS.
// Descriptor layout per CDNA5 ISA 8.3/8.4:
//   G0: count=1 | lds_addr | global_addr(57b) | type=2
//   G1: data_size=2 (4B), tensor_dim0=64, tensor_dim1=N, tile_dim0=64,
//       tile_dim1=TILE_ROWS, tensor_dim0_stride=64
// This toolchain exposes the 6-arg builtin:
//   (u32x4 g0, i32x8 g1, i32x4 g2, i32x4 g3, i32x8 extra, i32 cpol)
// ---------------------------------------------------------------------------
__device__ __forceinline__ void tdm_issue_tile(const float* base, int tile_idx,
                                               unsigned lds_addr, int n_rows) {
  uint64_t gaddr = (uint64_t)(uintptr_t)base +
                   (uint64_t)tile_idx * (TILE_ROWS * F_DIM * 4ull);
  u32x4 g0;
  g0[0] = 1u;                                            // count=1, no gather
  g0[1] = lds_addr;                                      // LDS byte address
  g0[2] = (unsigned)(gaddr & 0xFFFFFFFFull);             // global_addr[31:0]
  g0[3] = (unsigned)((gaddr >> 32) & 0x1FFFFFFull)       // global_addr[56:32]
        | (2u << 30);                                    // type = 2 ("image")
  i32x8 g1;
  g1[0] = (int)(2u << 16);                               // data_size = 4 bytes
  g1[1] = (int)(((unsigned)F_DIM & 0xFFFFu) << 16);      // tensor_dim0 lo16
  g1[2] = (int)((((unsigned)F_DIM >> 16) & 0xFFFFu)      // tensor_dim0 hi16
        | (((unsigned)n_rows & 0xFFFFu) << 16));         // tensor_dim1 lo16
  g1[3] = (int)((((unsigned)n_rows >> 16) & 0xFFFFu)     // tensor_dim1 hi16
        | ((unsigned)F_DIM << 16));                      // tile_dim0 = 64
  g1[4] = (int)(TILE_ROWS & 0xFFFF);                     // tile_dim1 (tile_dim2=0)
  g1[5] = (int)F_DIM;                                    // tensor_dim0_stride lo32
  g1[6] = 0;
  g1[7] = 0;
  i32x4 z4; z4[0] = 0; z4[1] = 0; z4[2] = 0; z4[3] = 0;  // groups 2/3 unused (2D)
  i32x8 z8; z8[0] = 0; z8[1] = 0; z8[2] = 0; z8[3] = 0;
  z8[4] = 0; z8[5] = 0; z8[6] = 0; z8[7] = 0;
  __builtin_amdgcn_tensor_load_to_lds(g0, g1, z4, z4, z8, 0);
}

// ---------------------------------------------------------------------------
// Kernel 2: scatter-push. 512 threads = 16 waves; one wave per source row of
// the current tile. Feature tiles are TDM double-buffered through LDS.
// Atomic adds land in the L2-resident 130MB accumulator.
// ---------------------------------------------------------------------------
__global__ void __launch_bounds__(512)
scatter_push_kernel(const float* __restrict__ features,
                    const float* __restrict__ wdown,
                    const int*   __restrict__ nidx,
                    float* __restrict__ denom,
                    float* __restrict__ feat_acc,
                    int n_rows, int n_tiles) {
  __shared__ float tile[2][TILE_ROWS * F_DIM];

  const int lane = threadIdx.x & 31;
  const int wave = threadIdx.x >> 5;          // 0..15 -> row within tile
  const bool issuer = (threadIdx.x < 32);     // wave 0 drives the TDM

  int cur = blockIdx.x;
  if (cur >= n_tiles) return;                 // uniform over the block

  if (issuer) {
    tdm_issue_tile(features, cur, (unsigned)(uintptr_t)&tile[0][0], n_rows);
  }

  int buf = 0;
  while (cur < n_tiles) {
    int nxt = cur + (int)gridDim.x;
    if (issuer) {
      if (nxt < n_tiles) {
        tdm_issue_tile(features, nxt, (unsigned)(uintptr_t)&tile[buf ^ 1][0],
                       n_rows);
        __builtin_amdgcn_s_wait_tensorcnt(1); // current tile landed in LDS
      } else {
        __builtin_amdgcn_s_wait_tensorcnt(0);
      }
    }
    __syncthreads();                          // tile[buf] visible to all waves

    int row = cur * TILE_ROWS + wave;
    if (row < n_rows) {
      // two feature columns per lane, straight out of LDS (conflict-free)
      float v0 = tile[buf][wave * F_DIM + lane];
      float v1 = tile[buf][wave * F_DIM + 32 + lane];

      // lanes 0..4 load the row's (weight, idx) pairs once; broadcast via shfl
      float wv = 0.f;
      int   jv = -1;
      if (lane < K_DIM) {
        wv = wdown[(long long)row * K_DIM + lane];
        jv = nidx [(long long)row * K_DIM + lane];
      }
#pragma unroll
      for (int k = 0; k < K_DIM; ++k) {
        float wk = __shfl(wv, k, 32);
        int   j  = __shfl(jv, k, 32);
        if (j >= 0) {                          // uniform across the wave
          if (lane == 0) {
            __hip_atomic_fetch_add(&denom[j], wk,
                                   __ATOMIC_RELAXED, __HIP_MEMORY_SCOPE_AGENT);
          }
          float* dst = feat_acc + (long long)j * F_DIM;
          __hip_atomic_fetch_add(dst + lane,      wk * v0,
                                 __ATOMIC_RELAXED, __HIP_MEMORY_SCOPE_AGENT);
          __hip_atomic_fetch_add(dst + 32 + lane, wk * v1,
                                 __ATOMIC_RELAXED, __HIP_MEMORY_SCOPE_AGENT);
        }
      }
    }
    __syncthreads();                          // everyone done reading tile[buf]
    buf ^= 1;
    cur = nxt;
  }
}

// ---------------------------------------------------------------------------
// Kernel 3: gather + normalize. One wave per output row, 2 columns per lane.
// ---------------------------------------------------------------------------
__global__ void __launch_bounds__(256)
gather_norm_kernel(const int*   __restrict__ sel,
                   const float* __restrict__ denom,
                   const float* __restrict__ feat_acc,
                   float* __restrict__ out, int n_up) {
  int lane = threadIdx.x & 31;
  int wave = threadIdx.x >> 5;
  int u = blockIdx.x * (blockDim.x >> 5) + wave;
  if (u >= n_up) return;

  int s = sel[u];
  float d = denom[s];
  const float* src = feat_acc + (long long)s * F_DIM;
  float r0 = (d == 0.f) ? 0.f : src[lane]      / d;
  float r1 = (d == 0.f) ? 0.f : src[32 + lane] / d;
  float* dst = out + (long long)u * F_DIM;
  dst[lane]      = r0;
  dst[32 + lane] = r1;
}

// ---------------------------------------------------------------------------
// Launch wrapper
// d_in: [0]=features (N*64 f32), [1]=weights_down (N*5 f32),
//       [2]=nidx_down (N*5 i32), [3]=sel_idx_up (N_UP i32)
// d_ws layout: denom[N] f32  |  feat_acc[N*64] f32   (~130 MB, fits in L2)
// ---------------------------------------------------------------------------
extern "C" void kernel_launch(void* const* d_in, const int* in_sizes, int n_in,
                              void* d_out, int out_size, void* d_ws,
                              size_t ws_size, hipStream_t stream) {
  const float* features = (const float*)d_in[0];
  const float* wdown    = (const float*)d_in[1];
  const int*   nidx     = (const int*)d_in[2];
  const int*   sel      = (const int*)d_in[3];

  const int N    = in_sizes[0] / F_DIM;
  const int N_UP = in_sizes[3];

  float* denom    = (float*)d_ws;
  float* feat_acc = denom + N;

  // 1) zero accumulators
  long long n_zero = (long long)N * (F_DIM + 1);
  zero_ws_kernel<<<2048, 256, 0, stream>>>(denom, n_zero);

  // 2) scatter with TDM-pipelined feature tiles
  int n_tiles = (N + TILE_ROWS - 1) / TILE_ROWS;
  int blocks  = n_tiles < 1024 ? n_tiles : 1024;
  scatter_push_kernel<<<blocks, 512, 0, stream>>>(features, wdown, nidx,
                                                  denom, feat_acc, N, n_tiles);

  // 3) gather + normalize
  int rows_per_block = 256 / 32;
  int gblocks = (N_UP + rows_per_block - 1) / rows_per_block;
  gather_norm_kernel<<<gblocks, 256, 0, stream>>>(sel, denom, feat_acc,
                                                  (float*)d_out, N_UP);
}